// NGCFLayer_52201032516155
// MI455X (gfx1250) — compile-verified
//
#include <hip/hip_runtime.h>
#include <math.h>

typedef __attribute__((ext_vector_type(2))) float v2f;
typedef __attribute__((ext_vector_type(8))) float v8f;
typedef unsigned int u32x4 __attribute__((ext_vector_type(4)));
typedef int i32x8 __attribute__((ext_vector_type(8)));
typedef int i32x4 __attribute__((ext_vector_type(4)));

#define D_FEAT 128

// ---------------------------------------------------------------------------
// Tensor Data Mover: DMA a 2D row-major f32 tile (rows x cols) from global
// memory into LDS. Descriptor fields per CDNA5 ISA.
// Issued by one wave; completion via s_wait_tensorcnt + workgroup barrier.
// This toolchain exposes the 6-arg builtin form (clang-23 / therock-10.0).
// ---------------------------------------------------------------------------
__device__ static inline void tdm_load_2d_f32(const float* gptr,
                                              unsigned lds_byte_off,
                                              unsigned rows, unsigned cols,
                                              unsigned row_stride_elems) {
  unsigned long long ga = (unsigned long long)(const void*)gptr;
  u32x4 g0;
  g0[0] = 1u;                                          // count=1, no gather
  g0[1] = lds_byte_off;                                // lds_addr [63:32]
  g0[2] = (unsigned)(ga & 0xffffffffu);                // global_addr[31:0]
  g0[3] = (unsigned)((ga >> 32) & 0x1ffffffu)          // global_addr[56:32]
          | (2u << 30);                                // type = 2 ("image")
  i32x8 g1;
  g1[0] = (int)(2u << 16);                             // data_size = 4 bytes
  g1[1] = (int)(cols << 16);                           // tensor_dim0[15:0]
  g1[2] = (int)(rows << 16);                           // tensor_dim1[15:0]
  g1[3] = (int)(cols << 16);                           // tile_dim0
  g1[4] = (int)rows;                                   // tile_dim1 (tile_dim2=0)
  g1[5] = (int)row_stride_elems;                       // tensor_dim0_stride[31:0]
  g1[6] = 0;                                           // strides high bits
  g1[7] = 0;
  i32x4 z4 = {0, 0, 0, 0};                             // 2D: groups 2/3 unused
  i32x8 z8 = {0, 0, 0, 0, 0, 0, 0, 0};
  __builtin_amdgcn_tensor_load_to_lds(g0, g1, z4, z4, z8, 0);
}

// ---------------------------------------------------------------------------
// Zero the workspace accumulators (grid-stride).
// ---------------------------------------------------------------------------
__global__ void zero_f32(float* __restrict__ p, long long n) {
  long long i = (long long)blockIdx.x * blockDim.x + threadIdx.x;
  long long stride = (long long)gridDim.x * blockDim.x;
  for (; i < n; i += stride) p[i] = 0.0f;
}

// ---------------------------------------------------------------------------
// Edge aggregation: one wave32 per edge, 4 floats per lane (float4).
//   agg1[dst] += norm * feat_src[src]
//   agg2[dst] += norm * (feat_src[src] * feat_dst[dst])
//   sums[dst] += norm
// Accumulators + feature tables fit in the 192 MB L2, so the f32 global
// atomics resolve at L2. Prefetch future edges' rows (global_prefetch_b8).
// ---------------------------------------------------------------------------
__global__ __launch_bounds__(256) void edge_agg(
    const float* __restrict__ feat_src, const float* __restrict__ feat_dst,
    const float* __restrict__ norm, const int* __restrict__ esrc,
    const int* __restrict__ edst, float* __restrict__ agg1,
    float* __restrict__ agg2, float* __restrict__ sums, int E) {
  int gid  = blockIdx.x * blockDim.x + threadIdx.x;
  int e    = gid >> 5;
  int lane = threadIdx.x & 31;
  if (e >= E) return;

  int   s  = esrc[e];
  int   d  = edst[e];
  float nv = norm[e];

  const float4* sp = (const float4*)(feat_src + (size_t)s * D_FEAT);
  const float4* dp = (const float4*)(feat_dst + (size_t)d * D_FEAT);
  float4 sv = sp[lane];   // 32 lanes x 4 floats = 128
  float4 dv = dp[lane];

  float* a1 = agg1 + (size_t)d * D_FEAT + lane * 4;
  float* a2 = agg2 + (size_t)d * D_FEAT + lane * 4;

  unsafeAtomicAdd(a1 + 0, nv * sv.x);
  unsafeAtomicAdd(a1 + 1, nv * sv.y);
  unsafeAtomicAdd(a1 + 2, nv * sv.z);
  unsafeAtomicAdd(a1 + 3, nv * sv.w);

  unsafeAtomicAdd(a2 + 0, nv * sv.x * dv.x);
  unsafeAtomicAdd(a2 + 1, nv * sv.y * dv.y);
  unsafeAtomicAdd(a2 + 2, nv * sv.z * dv.z);
  unsafeAtomicAdd(a2 + 3, nv * sv.w * dv.w);

  if (lane == 0) {
    unsafeAtomicAdd(sums + d, nv);
    int pf = e + 96;  // prefetch feature rows for an edge ~12 blocks ahead
    if (pf < E) {
      __builtin_prefetch(feat_src + (size_t)esrc[pf] * D_FEAT, 0, 3);
      __builtin_prefetch(feat_dst + (size_t)edst[pf] * D_FEAT, 0, 3);
    }
  }
}

// ---------------------------------------------------------------------------
// Per-node GEMM + epilogue, WMMA f32 16x16x4.
//   H = (feat + agg1) @ W1 + agg2 @ W2 + b1 + sums*(b1 + b2)
//   out = l2_normalize(leaky_relu(H, 0.2))
// One block = 16-row strip. 8 waves, each owns one 16x16 column tile.
// K = 256 fused loop over [W1 ; W2], DMA'd into LDS by the TDM (one
// tensor_load_to_lds per matrix, overlapped with cooperative A staging).
// A-strip (16 x 256) staged coalesced in LDS (16 KB); reused as the 16x128
// output staging buffer for the fused normalize epilogue.
// ---------------------------------------------------------------------------
__global__ __launch_bounds__(256) void node_gemm_finish(
    const float* __restrict__ feat, const float* __restrict__ agg1,
    const float* __restrict__ agg2, const float* __restrict__ W1,
    const float* __restrict__ W2, const float* __restrict__ b1,
    const float* __restrict__ b2, const float* __restrict__ sums,
    float* __restrict__ out, long long nRows) {
  extern __shared__ float smem[];
  float* sW = smem;                    // [256][128] : W1 rows then W2 rows
  float* sA = smem + 256 * D_FEAT;     // [16][256]  : A strip (later 16x128 H)

  const int tid  = threadIdx.x;
  const int lane = tid & 31;
  const int wv   = tid >> 5;                 // 0..7 -> column tile
  const long long rowBase = (long long)blockIdx.x * 16;

  // ---- TDM: DMA W1|W2 (2 x 64 KB) into LDS, issued by wave 0 -------------
  if (tid == 0) {
    unsigned offW = (unsigned)(unsigned long long)(void*)sW;
    tdm_load_2d_f32(W1, offW, D_FEAT, D_FEAT, D_FEAT);
    tdm_load_2d_f32(W2, offW + D_FEAT * D_FEAT * 4, D_FEAT, D_FEAT, D_FEAT);
  }

  // ---- stage A strip: row r, cols 0..127 = feat+agg1, 128..255 = agg2 ----
  {
    float4* sAv = (float4*)sA;
    for (int i = tid; i < (16 * 256) / 4; i += 256) {
      int r  = i >> 6;        // 64 float4 per row
      int c4 = i & 63;
      long long node = rowBase + r;
      float4 v = {0.f, 0.f, 0.f, 0.f};
      if (node < nRows) {
        if (c4 < 32) {
          float4 a = ((const float4*)(feat + node * D_FEAT))[c4];
          float4 b = ((const float4*)(agg1 + node * D_FEAT))[c4];
          v = float4{a.x + b.x, a.y + b.y, a.z + b.z, a.w + b.w};
        } else {
          v = ((const float4*)(agg2 + node * D_FEAT))[c4 - 32];
        }
      }
      sAv[i] = v;
    }
  }
  if (tid == 0) {
    __builtin_amdgcn_s_wait_tensorcnt(0);   // TDM writes to sW complete
  }
  __syncthreads();

  // ---- WMMA main loop: C(16x16) += A(16x4) * B(4x16), K = 256 ------------
  // A frag: lanes 0-15 hold row m, k+0/k+1; lanes 16-31 hold row m, k+2/k+3.
  // B frag mirrors A: lanes 0-15 col n rows k+0/k+1; lanes 16-31 rows k+2/3.
  v8f c = {};
  const int m    = lane & 15;
  const int koff = (lane >> 4) << 1;   // 0 or 2
  const int n    = wv * 16 + (lane & 15);

  for (int k = 0; k < 256; k += 4) {
    float2 av = *(const float2*)(sA + m * 256 + k + koff);   // ds_load_b64
    v2f a; a.x = av.x; a.y = av.y;
    v2f b;
    b.x = sW[(k + koff + 0) * D_FEAT + n];
    b.y = sW[(k + koff + 1) * D_FEAT + n];
    c = __builtin_amdgcn_wmma_f32_16x16x4_f32(
        /*neg_a=*/false, a, /*neg_b=*/false, b,
        /*c_mod=*/(short)0, c, /*reuse_a=*/false, /*reuse_b=*/false);
  }
  __syncthreads();   // all waves done reading sA before it is reused for H

  // ---- write C + bias into LDS H buffer (reuse sA: 16x128) ---------------
  float* sH = sA;
  {
    const float bn1 = b1[n];
    const float bnb = bn1 + b2[n];
    #pragma unroll
    for (int i = 0; i < 8; ++i) {
      int r = (lane < 16) ? i : (i + 8);        // C layout: VGPR i -> M=i / M=8+i
      long long node = rowBase + r;
      float sm = (node < nRows) ? sums[node] : 0.f;
      sH[r * D_FEAT + n] = c[i] + bn1 + sm * bnb;
    }
  }
  __syncthreads();

  // ---- fused epilogue: leaky_relu(0.2) + row L2 normalize ----------------
  {
    int r    = tid >> 4;          // 0..15
    int cseg = (tid & 15) * 8;    // 8 columns per thread
    float v[8];
    float ss = 0.f;
    #pragma unroll
    for (int j = 0; j < 8; ++j) {
      float x = sH[r * D_FEAT + cseg + j];
      x = (x >= 0.f) ? x : 0.2f * x;
      v[j] = x;
      ss += x * x;
    }
    // reduce across the 16 lanes owning this row (groups align to half-wave)
    ss += __shfl_xor(ss, 8, 16);
    ss += __shfl_xor(ss, 4, 16);
    ss += __shfl_xor(ss, 2, 16);
    ss += __shfl_xor(ss, 1, 16);
    float nrm   = sqrtf(ss);
    float scale = 1.0f / fmaxf(nrm, 1e-12f);
    long long node = rowBase + r;
    if (node < nRows) {
      float* op = out + node * D_FEAT + cseg;
      #pragma unroll
      for (int j = 0; j < 8; ++j) op[j] = v[j] * scale;
    }
  }
}

// ---------------------------------------------------------------------------
extern "C" void kernel_launch(void* const* d_in, const int* in_sizes, int n_in,
                              void* d_out, int out_size, void* d_ws, size_t ws_size,
                              hipStream_t stream) {
  const float* feat_user = (const float*)d_in[0];
  const float* feat_item = (const float*)d_in[1];
  const float* W1        = (const float*)d_in[2];
  const float* b1        = (const float*)d_in[3];
  const float* W2        = (const float*)d_in[4];
  const float* b2        = (const float*)d_in[5];
  const float* norm_u2i  = (const float*)d_in[6];
  const float* norm_i2u  = (const float*)d_in[7];
  const int* e_u2i_src   = (const int*)d_in[8];
  const int* e_u2i_dst   = (const int*)d_in[9];
  const int* e_i2u_src   = (const int*)d_in[10];
  const int* e_i2u_dst   = (const int*)d_in[11];
  float* out = (float*)d_out;

  const int D = in_sizes[3];                    // 128
  const long long NU = in_sizes[0] / D;         // 100000
  const long long NI = in_sizes[1] / D;         // 50000
  const int E = in_sizes[8];                    // 500000

  // workspace: aggU1, aggU2, aggI1, aggI2 (D each per node), sU, sI
  float* ws    = (float*)d_ws;
  float* aggU1 = ws;
  float* aggU2 = aggU1 + NU * D;
  float* aggI1 = aggU2 + NU * D;
  float* aggI2 = aggI1 + NI * D;
  float* sU    = aggI2 + NI * D;
  float* sI    = sU + NU;
  long long zn = 2 * (NU + NI) * D + NU + NI;

  zero_f32<<<2048, 256, 0, stream>>>(ws, zn);

  int eb = (E + 7) / 8;  // 8 edges per 256-thread block (one wave32 each)
  edge_agg<<<eb, 256, 0, stream>>>(feat_user, feat_item, norm_u2i,
                                   e_u2i_src, e_u2i_dst, aggI1, aggI2, sI, E);
  edge_agg<<<eb, 256, 0, stream>>>(feat_item, feat_user, norm_i2u,
                                   e_i2u_src, e_i2u_dst, aggU1, aggU2, sU, E);

  size_t lds = (size_t)(256 * D_FEAT + 16 * 256) * sizeof(float);  // 147456 B
  int gbU = (int)((NU + 15) / 16);
  int gbI = (int)((NI + 15) / 16);
  node_gemm_finish<<<gbU, 256, lds, stream>>>(feat_user, aggU1, aggU2, W1, W2,
                                              b1, b2, sU, out, NU);
  node_gemm_finish<<<gbI, 256, lds, stream>>>(feat_item, aggI1, aggI2, W1, W2,
                                              b1, b2, sI, out + NU * D, NI);
}